// OptimalTransportMatcher_28192165331076
// MI455X (gfx1250) — compile-verified
//
#include <hip/hip_runtime.h>
#include <hip/hip_bf16.h>

// OptimalTransportMatcher (SuperGlue-style Sinkhorn) for MI455X / gfx1250.
//
// Math: with s0 = augmented scores (dustbin d in row N / col M), A = 2*s0,
//   q = collse(s0)                      (initial row-lse cancels out)
//   repeat 5x: p_r = lse_c(A - q_c);  q_c = lse_r(A - p_r)
//   out = exp(A - p - q)[:N,:M]
// Only per-row / per-col scalars are updated between passes; the 134 MB score
// matrix is written once and stays resident in the 192 MB L2 for all passes.
//
// Requires ws_size >= ~137 MB (scores + potentials + column partials).

typedef float v2f __attribute__((ext_vector_type(2)));
typedef float v8f __attribute__((ext_vector_type(8)));

namespace otm {
constexpr int B = 32, N = 1024, M = 1024, D = 256;
constexpr int NP1 = N + 1, MP1 = M + 1;
constexpr int TM = 32, TN = 64, KC = 64;   // GEMM tile: 32x64, K-chunk 64
constexpr int RC = 8;                      // row chunks for column reduction
constexpr int RCH = (NP1 + RC - 1) / RC;   // 129 rows per chunk
constexpr float NEG_HUGE = -3.402823466e+38f;

__device__ __forceinline__ void lse_join(float& m1, float& s1, float m2, float s2) {
  float m = fmaxf(m1, m2);
  s1 = s1 * __expf(m1 - m) + s2 * __expf(m2 - m);
  m1 = m;
}
__device__ __forceinline__ void lse_push(float& mx, float& sm, float x) {
  float m = fmaxf(mx, x);
  sm = sm * __expf(mx - m) + __expf(x - m);
  mx = m;
}
} // namespace otm

// ---------------------------------------------------------------------------
// Kernel 1: batched GEMM  s[b,r,c] = sum_k desc1[b,r,k] * desc2[b,c,k]
// Block = 256 threads (8 waves). Tile 32(rows) x 64(cols); each wave owns one
// 16x16 subtile and issues V_WMMA_F32_16X16X4_F32 over K.
// ---------------------------------------------------------------------------
__global__ __launch_bounds__(256)
void otm_gemm_wmma(const float* __restrict__ d1, const float* __restrict__ d2,
                   float* __restrict__ s) {
  using namespace otm;
  __shared__ float As[TM][KC + 4];   // stride 68: banks (4*row + k) % 64, conflict-free
  __shared__ float Bs[TN][KC + 4];

  const int tid  = threadIdx.x;
  const int lane = tid & 31;
  const int wave = tid >> 5;
  const int wr16 = (wave & 1) << 4;  // 0 / 16 : row offset of subtile
  const int wc16 = (wave >> 1) << 4; // 0..48  : col offset of subtile
  const int r0 = blockIdx.y * TM;
  const int c0 = blockIdx.x * TN;
  const int bz = blockIdx.z;

  // ISA 7.12.2: 32-bit A 16x4 -> lane L holds row m = L&15, K pair
  // {0,1} (lanes 0-15) or {2,3} (lanes 16-31). B 4x16 mirrors with n = L&15.
  const int m   = lane & 15;
  const int klo = (lane >> 4) << 1;

  const float* a_base = d1 + ((size_t)bz * N + r0) * D;
  const float* b_base = d2 + ((size_t)bz * M + c0) * D;

  v8f acc = {};

  for (int kc = 0; kc < D; kc += KC) {
    // Stage A tile: 32 rows x 64 K (512 float4, 2 per thread)
    for (int i = tid; i < TM * (KC / 4); i += 256) {
      const int row = i >> 4, k4 = (i & 15) << 2;
      const float4 v = *(const float4*)(a_base + (size_t)row * D + kc + k4);
      As[row][k4 + 0] = v.x; As[row][k4 + 1] = v.y;
      As[row][k4 + 2] = v.z; As[row][k4 + 3] = v.w;
    }
    // Stage B tile: 64 cols x 64 K (1024 float4, 4 per thread)
    for (int i = tid; i < TN * (KC / 4); i += 256) {
      const int row = i >> 4, k4 = (i & 15) << 2;
      const float4 v = *(const float4*)(b_base + (size_t)row * D + kc + k4);
      Bs[row][k4 + 0] = v.x; Bs[row][k4 + 1] = v.y;
      Bs[row][k4 + 2] = v.z; Bs[row][k4 + 3] = v.w;
    }
    __syncthreads();

#pragma unroll
    for (int k = 0; k < KC; k += 4) {
      v2f a, b;
      a.x = As[wr16 + m][k + klo];
      a.y = As[wr16 + m][k + klo + 1];
      b.x = Bs[wc16 + m][k + klo];
      b.y = Bs[wc16 + m][k + klo + 1];
      acc = __builtin_amdgcn_wmma_f32_16x16x4_f32(
          /*neg_a=*/false, a, /*neg_b=*/false, b,
          /*c_mod=*/(short)0, acc, /*reuse_a=*/false, /*reuse_b=*/false);
    }
    __syncthreads();
  }

  // D layout: VGPR v -> rows v (lanes 0-15) / v+8 (lanes 16-31), col = lane&15
  const int rowhi = (lane >> 4) << 3;
  float* out = s + ((size_t)bz * N + r0 + wr16 + rowhi) * M + c0 + wc16 + (lane & 15);
#pragma unroll
  for (int v = 0; v < 8; ++v) {
    out[(size_t)v * M] = acc[v];
  }
}

// ---------------------------------------------------------------------------
// Kernel 2: p[b,r] = lse_c( alpha * s0[b,r,c] - q[b,c] ),  r in [0, N]
// One block per (row, batch). Dustbin row/col value = alpha * d.
// ---------------------------------------------------------------------------
__global__ __launch_bounds__(256)
void otm_row_lse(const float* __restrict__ s, const float* __restrict__ q,
                 const float* __restrict__ dustbin, float* __restrict__ p,
                 float alpha) {
  using namespace otm;
  __shared__ float smx[256], ssm[256];
  const int tid = threadIdx.x;
  const int r = blockIdx.x;
  const int b = blockIdx.y;
  const float ad = alpha * dustbin[0];
  const float* qb = q + (size_t)b * MP1;

  float mx = NEG_HUGE, sm = 0.0f;
  if (r < N) {
    const float* row = s + ((size_t)b * N + r) * M;
    for (int c = tid; c < M; c += 256)
      lse_push(mx, sm, alpha * row[c] - qb[c]);
    if (tid == 0) lse_push(mx, sm, ad - qb[M]);   // dustbin column
  } else {
    for (int c = tid; c < MP1; c += 256)          // dustbin row: all entries = d
      lse_push(mx, sm, ad - qb[c]);
  }

  smx[tid] = mx; ssm[tid] = sm;
  __syncthreads();
  for (int off = 128; off > 0; off >>= 1) {
    if (tid < off) lse_join(smx[tid], ssm[tid], smx[tid + off], ssm[tid + off]);
    __syncthreads();
  }
  if (tid == 0) p[(size_t)b * NP1 + r] = smx[0] + __logf(ssm[0]);
}

// ---------------------------------------------------------------------------
// Kernel 3a: partial column lse over a chunk of rows.
// thread = one column (coalesced across the row sweep). p may be null (init).
// part[(b*RC + chunk)*MP1 + c] = {running max, running sum}
// ---------------------------------------------------------------------------
__global__ __launch_bounds__(256)
void otm_col_lse_part(const float* __restrict__ s, const float* __restrict__ p,
                      const float* __restrict__ dustbin,
                      float2* __restrict__ part, float alpha) {
  using namespace otm;
  const int c = blockIdx.x * 256 + threadIdx.x;
  if (c >= MP1) return;
  const int b = blockIdx.z;
  const int rbeg = blockIdx.y * RCH;
  const int rend = min(rbeg + RCH, NP1);
  const float ad = alpha * dustbin[0];
  const float* pb = p ? (p + (size_t)b * NP1) : nullptr;

  float mx = NEG_HUGE, sm = 0.0f;
  for (int r = rbeg; r < rend; ++r) {
    float v = (r < N && c < M) ? alpha * s[((size_t)b * N + r) * M + c] : ad;
    if (pb) v -= pb[r];
    lse_push(mx, sm, v);
  }
  part[((size_t)b * RC + blockIdx.y) * MP1 + c] = make_float2(mx, sm);
}

// Kernel 3b: combine RC partials -> q[b,c] = mx + log(sm)
__global__ __launch_bounds__(256)
void otm_col_lse_comb(const float2* __restrict__ part, float* __restrict__ q) {
  using namespace otm;
  const int c = blockIdx.x * 256 + threadIdx.x;
  if (c >= MP1) return;
  const int b = blockIdx.y;
  float mx = NEG_HUGE, sm = 0.0f;
  for (int ch = 0; ch < RC; ++ch) {
    const float2 v = part[((size_t)b * RC + ch) * MP1 + c];
    lse_join(mx, sm, v.x, v.y);
  }
  q[(size_t)b * MP1 + c] = mx + __logf(sm);
}

// ---------------------------------------------------------------------------
// Kernel 4: out[b,r,c] = exp(2*s - p_r - q_c), float4-vectorized.
// ---------------------------------------------------------------------------
__global__ __launch_bounds__(256)
void otm_finalize(const float* __restrict__ s, const float* __restrict__ p,
                  const float* __restrict__ q, float* __restrict__ out) {
  using namespace otm;
  const size_t idx = (size_t)blockIdx.x * 256 + threadIdx.x;   // over B*N*(M/4)
  const int b = (int)(idx / ((size_t)N * (M / 4)));
  const size_t rem = idx % ((size_t)N * (M / 4));
  const int r = (int)(rem / (M / 4));
  const int c4 = (int)(rem % (M / 4)) << 2;

  const float4 sv = *(const float4*)(s + ((size_t)b * N + r) * M + c4);
  const float pr = p[(size_t)b * NP1 + r];
  const float* qb = q + (size_t)b * MP1 + c4;
  float4 o;
  o.x = __expf(2.0f * sv.x - pr - qb[0]);
  o.y = __expf(2.0f * sv.y - pr - qb[1]);
  o.z = __expf(2.0f * sv.z - pr - qb[2]);
  o.w = __expf(2.0f * sv.w - pr - qb[3]);
  *(float4*)(out + ((size_t)b * N + r) * M + c4) = o;
}

// ---------------------------------------------------------------------------
extern "C" void kernel_launch(void* const* d_in, const int* in_sizes, int n_in,
                              void* d_out, int out_size, void* d_ws, size_t ws_size,
                              hipStream_t stream) {
  using namespace otm;
  const float* desc1 = (const float*)d_in[0];
  const float* desc2 = (const float*)d_in[1];
  const float* dust  = (const float*)d_in[2];
  float* out = (float*)d_out;

  // Workspace layout (floats): scores | p | q | column partials
  constexpr size_t S_ELEMS = (size_t)B * N * M;     // 33,554,432
  constexpr size_t PQ = (size_t)B * NP1;            // 32,800
  float*  s    = (float*)d_ws;
  float*  p    = s + S_ELEMS;
  float*  q    = p + PQ;
  float2* part = (float2*)(q + PQ);                 // B * RC * MP1 float2

  constexpr int NUM_ITERATIONS = 5;
  const dim3 blk(256);

  // 1) scores = desc1 . desc2^T  (WMMA f32)
  otm_gemm_wmma<<<dim3(M / TN, N / TM, B), blk, 0, stream>>>(desc1, desc2, s);

  // 2) q = collse(s0)  (alpha = 1, no potential)
  otm_col_lse_part<<<dim3((MP1 + 255) / 256, RC, B), blk, 0, stream>>>(
      s, nullptr, dust, part, 1.0f);
  otm_col_lse_comb<<<dim3((MP1 + 255) / 256, B), blk, 0, stream>>>(part, q);

  // 3) Sinkhorn iterations on A = 2*s0
  for (int it = 0; it < NUM_ITERATIONS; ++it) {
    otm_row_lse<<<dim3(NP1, B), blk, 0, stream>>>(s, q, dust, p, 2.0f);
    otm_col_lse_part<<<dim3((MP1 + 255) / 256, RC, B), blk, 0, stream>>>(
        s, p, dust, part, 2.0f);
    otm_col_lse_comb<<<dim3((MP1 + 255) / 256, B), blk, 0, stream>>>(part, q);
  }

  // 4) out = exp(2*s - p - q)[:N,:M]
  otm_finalize<<<dim3((unsigned)((size_t)B * N * (M / 4) / 256)), blk, 0, stream>>>(
      s, p, q, out);
}